// BispectrumCalculator_2190433320997
// MI455X (gfx1250) — compile-verified
//
#include <hip/hip_runtime.h>

#define NN 512
#define NMASK 511

typedef __attribute__((ext_vector_type(2))) float v2f;
typedef __attribute__((ext_vector_type(8))) float v8f;

// ---------------------------------------------------------------------------
// Kernel 1: dense DFT of 128 rows x 512 via V_WMMA_F32_16X16X4_F32.
// Block = 256 threads = 8 waves. Block handles one 16-row tile of X (shared in
// LDS) and 8 consecutive 16-col frequency tiles (one per wave).
// yr[f] = sum_j x[j] cos(2*pi*j*f/N);  yi[f] = -sum_j x[j] sin(2*pi*j*f/N)
// ---------------------------------------------------------------------------
__global__ __launch_bounds__(256) void dft_wmma_kernel(const float* __restrict__ x,
                                                       float* __restrict__ yr,
                                                       float* __restrict__ yi) {
    __shared__ float cosT[NN];
    __shared__ float sinT[NN];
    __shared__ float sx[16][NN + 1];   // +1 pad: stride 513 avoids bank conflicts

    const int tid = threadIdx.x;
    const int mt  = blockIdx.x >> 2;   // row tile   0..7   (16 rows each)
    const int ftg = blockIdx.x & 3;    // freq group 0..3   (8 tiles each)

    const float w0 = 6.283185307179586f / (float)NN;
    for (int i = tid; i < NN; i += 256) {
        float ang = w0 * (float)i;
        cosT[i] = __builtin_cosf(ang);
        sinT[i] = __builtin_sinf(ang);
    }
    for (int i = tid; i < 16 * NN; i += 256) {
        int row = i >> 9, col = i & NMASK;
        sx[row][col] = x[(mt * 16 + row) * NN + col];
    }
    __syncthreads();

    const int wave = tid >> 5;
    const int lane = tid & 31;
    const int ft   = ftg * 8 + wave;        // frequency tile 0..31
    const int m    = lane & 15;             // A: M index / B,D: N index
    const int jh   = (lane >> 4) << 1;      // K sub-offset (0 or 2)
    const int f    = ft * 16 + m;           // global frequency for B fragment

    v8f accR = {};
    v8f accI = {};
    for (int j0 = 0; j0 < NN; j0 += 4) {
        const int j = j0 + jh;
        v2f a, bc, bs;
        a.x = sx[m][j];
        a.y = sx[m][j + 1];
        const int p0 = (j * f) & NMASK;
        const int p1 = ((j + 1) * f) & NMASK;
        bc.x = cosT[p0]; bc.y = cosT[p1];
        bs.x = sinT[p0]; bs.y = sinT[p1];
        accR = __builtin_amdgcn_wmma_f32_16x16x4_f32(false, a, false, bc,
                                                     (short)0, accR, false, false);
        accI = __builtin_amdgcn_wmma_f32_16x16x4_f32(false, a, false, bs,
                                                     (short)0, accI, false, false);
    }

    // D layout: VGPR v -> M = 8*(lane>=16) + v, N = lane&15
    const int fOut  = ft * 16 + (lane & 15);
    const int mBase = mt * 16 + ((lane >> 4) << 3);
#pragma unroll
    for (int v = 0; v < 8; ++v) {
        const int row = mBase + v;
        yr[row * NN + fOut] =  accR[v];
        yi[row * NN + fOut] = -accI[v];   // fft uses exp(-i theta)
    }
}

// ---------------------------------------------------------------------------
// Kernel 2 (bandwidth kernel): Bx[k,l] = y[k]*conj(y[l])*y[(l-k)&511], mean
// over t=0..3, real->ch0, imag->ch1. One block = fixed b, 4 k-rows, all 512 l.
// y[b] staged in LDS (16 KB). Interleaved l across threads: conflict-free LDS
// reads and fully coalesced 128 B global stores per wave.
// ---------------------------------------------------------------------------
__global__ __launch_bounds__(256) void bispectrum_kernel(const float* __restrict__ yr,
                                                         const float* __restrict__ yi,
                                                         float* __restrict__ out) {
    const int b  = blockIdx.x >> 7;    // 0..31
    const int kt = blockIdx.x & 127;   // 0..127, 4 k-rows each

    __shared__ float syr[4][NN];
    __shared__ float syi[4][NN];
    for (int i = threadIdx.x; i < 4 * NN; i += 256) {
        int t = i >> 9, j = i & NMASK;
        syr[t][j] = yr[(b * 4 + t) * NN + j];
        syi[t][j] = yi[(b * 4 + t) * NN + j];
    }
    __syncthreads();

    const int kl  = threadIdx.x >> 6;      // local k row 0..3
    const int l64 = threadIdx.x & 63;      // interleave base
    const int k   = kt * 4 + kl;

    float kre[4], kim[4];
#pragma unroll
    for (int t = 0; t < 4; ++t) { kre[t] = syr[t][k]; kim[t] = syi[t][k]; }

    const size_t base0 = ((size_t)(b * 2 + 0) * NN + (size_t)k) * NN;
    const size_t base1 = ((size_t)(b * 2 + 1) * NN + (size_t)k) * NN;

#pragma unroll
    for (int li = 0; li < 8; ++li) {
        const int l  = l64 + (li << 6);
        const int mm = (l - k) & NMASK;
        float sr = 0.0f, si = 0.0f;
#pragma unroll
        for (int t = 0; t < 4; ++t) {
            const float lr = syr[t][l],  lq = syi[t][l];
            const float mr = syr[t][mm], mi = syi[t][mm];
            // p = y[k] * conj(y[l])
            const float pr = kre[t] * lr + kim[t] * lq;
            const float pi = kim[t] * lr - kre[t] * lq;
            // q = p * y[(l-k) mod N], accumulate
            sr += pr * mr - pi * mi;
            si += pr * mi + pi * mr;
        }
        out[base0 + l] = sr * 0.25f;
        out[base1 + l] = si * 0.25f;
    }
}

extern "C" void kernel_launch(void* const* d_in, const int* in_sizes, int n_in,
                              void* d_out, int out_size, void* d_ws, size_t ws_size,
                              hipStream_t stream) {
    (void)in_sizes; (void)n_in; (void)out_size; (void)ws_size;
    const float* target = (const float*)d_in[0];       // [32,4,512] f32
    float* out = (float*)d_out;                        // [32,2,512,512] + [32,4,512]
    float* yr  = (float*)d_ws;                         // [128][512]
    float* yi  = yr + 128 * NN;                        // [128][512]  (512 KB total)

    dft_wmma_kernel<<<32, 256, 0, stream>>>(target, yr, yi);
    bispectrum_kernel<<<32 * 128, 256, 0, stream>>>(yr, yi, out);

    // tuple's second output: pass-through of target
    hipMemcpyAsync(out + (size_t)32 * 2 * NN * NN, target,
                   (size_t)128 * NN * sizeof(float),
                   hipMemcpyDeviceToDevice, stream);
}